// LightGCN_37091337568955
// MI455X (gfx1250) — compile-verified
//
#include <hip/hip_runtime.h>
#include <stdint.h>

#define EDIM   64
#define SCALE  0.25f     // 1/(NUM_LAYERS+1)
#define TPB    256
#define WPB    8         // waves per block (wave32)
#define BATCH  32        // edges staged per wave per iteration

// ---------- CDNA5 async global->LDS staging (ASYNCcnt path) ----------------
#if __has_builtin(__builtin_amdgcn_global_load_async_to_lds_b32)
#define ASYNC_CP_B32(gsrc, ldst)                                              \
  __builtin_amdgcn_global_load_async_to_lds_b32(                              \
      (__attribute__((address_space(1))) int*)(uintptr_t)(gsrc),              \
      (__attribute__((address_space(3))) int*)(uint32_t)(uintptr_t)(ldst),    \
      0, 0)
#else
#define ASYNC_CP_B32(gsrc, ldst)                                              \
  asm volatile("global_load_async_to_lds_b32 %0, %1, off"                     \
               :: "v"((uint32_t)(uintptr_t)(ldst)),                           \
                  "v"((uint64_t)(uintptr_t)(gsrc))                            \
               : "memory")
#endif

#if __has_builtin(__builtin_amdgcn_s_wait_asynccnt)
#define WAIT_ASYNC(n) __builtin_amdgcn_s_wait_asynccnt(n)
#else
#define WAIT_ASYNC(n) asm volatile("s_wait_asynccnt %0" :: "i"(n) : "memory")
#endif

// ---------------------------------------------------------------------------
__global__ void lgcn_init(float4* __restrict__ out4, const float4* __restrict__ emb4,
                          float4* __restrict__ h4, float* __restrict__ deg,
                          int n4, int N)
{
  int i = blockIdx.x * blockDim.x + threadIdx.x;
  if (i < n4) {
    float4 e = emb4[i];
    out4[i] = make_float4(e.x * SCALE, e.y * SCALE, e.z * SCALE, e.w * SCALE);
    h4[i]   = make_float4(0.f, 0.f, 0.f, 0.f);
  }
  if (i < N) deg[i] = 0.0f;
}

__global__ void lgcn_degree(const int* __restrict__ rows, float* __restrict__ deg, int E)
{
  int e = blockIdx.x * blockDim.x + threadIdx.x;
  if (e < E) atomicAdd(&deg[rows[e]], 1.0f);
}

__global__ void lgcn_dinv(const float* __restrict__ deg, float* __restrict__ dinv, int N)
{
  int i = blockIdx.x * blockDim.x + threadIdx.x;
  if (i < N) {
    float d = deg[i];
    dinv[i] = (d > 0.0f) ? rsqrtf(d) : 0.0f;
  }
}

// One wave processes BATCH=32 edges per iteration. Edge indices for the next
// batch are prefetched into a wave-private LDS double buffer with async
// global->LDS copies. Inside a batch: lanes 0-15 handle edge j, lanes 16-31
// handle edge j+1; each lane owns one float4 (16B) of the 256B embedding row,
// so gathers are fully coalesced and each lane issues 4 fp32 atomics.
__global__ __launch_bounds__(TPB) void lgcn_scatter(
    const int* __restrict__ rows, const int* __restrict__ cols,
    const float* __restrict__ dinv,
    const float* __restrict__ hprev, float* __restrict__ hnext, int E)
{
  __shared__ int   s_row[WPB][2][BATCH];
  __shared__ int   s_col[WPB][2][BATCH];
  __shared__ float s_nrm[WPB][BATCH];

  const int lane   = threadIdx.x & 31;
  const int wslot  = threadIdx.x >> 5;
  const int nWaves = (gridDim.x * TPB) >> 5;
  const int wid    = (blockIdx.x * TPB + threadIdx.x) >> 5;
  const int nBatch = (E + BATCH - 1) / BATCH;

  if (wid >= nBatch) return;

  auto issue = [&](int b, int half) {
    int e = b * BATCH + lane;
    if (e >= E) e = E - 1;                    // clamp; tail guarded at use
    ASYNC_CP_B32(rows + e, &s_row[wslot][half][lane]);
    ASYNC_CP_B32(cols + e, &s_col[wslot][half][lane]);
  };

  issue(wid, 0);

  const int sub = lane >> 4;          // which of the 2 concurrent edges
  const int d4  = (lane & 15) << 2;   // starting feature dim of this lane's float4

  int k = 0;
  for (int b = wid; b < nBatch; b += nWaves, k ^= 1) {
    int nb = b + nWaves;
    if (nb < nBatch) { issue(nb, k ^ 1); WAIT_ASYNC(2); }  // batch b is done (in-order)
    else             { WAIT_ASYNC(0); }
    asm volatile("" ::: "memory");

    // Phase 1: one edge per lane -> cache its norm in LDS
    {
      int r = s_row[wslot][k][lane];
      int c = s_col[wslot][k][lane];
      s_nrm[wslot][lane] = dinv[r] * dinv[c];
    }
    __builtin_amdgcn_wave_barrier();

    // Phase 2: cooperative gather/scale/scatter, 2 edges at a time
    const int base = b * BATCH;
    #pragma unroll 4
    for (int j = 0; j < BATCH; j += 2) {
      int ei = j + sub;
      if (base + ei < E) {
        int   r   = s_row[wslot][k][ei];
        int   c   = s_col[wslot][k][ei];
        float nrm = s_nrm[wslot][ei];
        const float4 v = *reinterpret_cast<const float4*>(hprev + (size_t)r * EDIM + d4);
        float* dst = hnext + (size_t)c * EDIM + d4;
        atomicAdd(dst + 0, v.x * nrm);
        atomicAdd(dst + 1, v.y * nrm);
        atomicAdd(dst + 2, v.z * nrm);
        atomicAdd(dst + 3, v.w * nrm);
      }
    }
    __builtin_amdgcn_wave_barrier();
  }
}

__global__ void lgcn_acc(float4* __restrict__ out4, const float4* __restrict__ h4,
                         float4* __restrict__ zero4, int n4)
{
  int i = blockIdx.x * blockDim.x + threadIdx.x;
  if (i < n4) {
    float4 o = out4[i], h = h4[i];
    o.x += h.x * SCALE; o.y += h.y * SCALE;
    o.z += h.z * SCALE; o.w += h.w * SCALE;
    out4[i]  = o;
    zero4[i] = make_float4(0.f, 0.f, 0.f, 0.f);   // prep next ping-pong buffer
  }
}

// ---------------------------------------------------------------------------
extern "C" void kernel_launch(void* const* d_in, const int* in_sizes, int n_in,
                              void* d_out, int out_size, void* d_ws, size_t ws_size,
                              hipStream_t stream) {
  const int*   eidx = (const int*)d_in[0];     // edge_index [2, E] (int32)
  const float* emb  = (const float*)d_in[1];   // embeddings [N, 64] (f32)
  const int E = in_sizes[0] / 2;
  const int N = in_sizes[1] / EDIM;
  const int n4 = (N * EDIM) / 4;

  const int* rows = eidx;
  const int* cols = eidx + E;

  // Workspace layout: deg[N] | dinv[N] | hA[N*64] | hB[N*64]  (~78 MB)
  float* ws   = (float*)d_ws;
  float* deg  = ws;
  float* dinv = ws + (size_t)N;
  float* hA   = ws + 2 * (size_t)N;
  float* hB   = hA + (size_t)N * EDIM;
  float* out  = (float*)d_out;

  const int bE  = (E + TPB - 1) / TPB;
  const int bN  = (N + TPB - 1) / TPB;
  const int b4  = (n4 + TPB - 1) / TPB;
  const int bSc = 768;   // 768 blocks * 8 waves = 6144 waves over 62500 batches

  // out = emb*scale; hA = 0; deg = 0
  lgcn_init<<<b4, TPB, 0, stream>>>((float4*)out, (const float4*)emb,
                                    (float4*)hA, deg, n4, N);
  // deg[row] += 1
  lgcn_degree<<<bE, TPB, 0, stream>>>(rows, deg, E);
  // dinv = deg>0 ? rsqrt(deg) : 0
  lgcn_dinv<<<bN, TPB, 0, stream>>>(deg, dinv, N);

  // Layer 1: emb -> hA ; out += hA*s ; zero hB
  lgcn_scatter<<<bSc, TPB, 0, stream>>>(rows, cols, dinv, emb, hA, E);
  lgcn_acc<<<b4, TPB, 0, stream>>>((float4*)out, (const float4*)hA, (float4*)hB, n4);

  // Layer 2: hA -> hB ; out += hB*s ; zero hA
  lgcn_scatter<<<bSc, TPB, 0, stream>>>(rows, cols, dinv, hA, hB, E);
  lgcn_acc<<<b4, TPB, 0, stream>>>((float4*)out, (const float4*)hB, (float4*)hA, n4);

  // Layer 3: hB -> hA ; out += hA*s ; zero hB (harmless)
  lgcn_scatter<<<bSc, TPB, 0, stream>>>(rows, cols, dinv, hB, hA, E);
  lgcn_acc<<<b4, TPB, 0, stream>>>((float4*)out, (const float4*)hA, (float4*)hB, n4);
}